// GraphClassifier_3401614098982
// MI455X (gfx1250) — compile-verified
//
#include <hip/hip_runtime.h>

#define NNODES  50000
#define NEDGES  800000
#define NGRAPHS 64
#define FIN  128
#define FHID 128
#define FOUT 16

typedef __attribute__((ext_vector_type(2))) float v2f;
typedef __attribute__((ext_vector_type(8))) float v8f;

// ---------------- utility kernels ----------------

__global__ __launch_bounds__(256) void k_zero(float* __restrict__ p, int n) {
    int i = blockIdx.x * 256 + threadIdx.x;
    if (i < n) p[i] = 0.0f;
}

__global__ __launch_bounds__(256) void k_degree(const int* __restrict__ src,
                                                const int* __restrict__ dst,
                                                float* __restrict__ deg_out,
                                                float* __restrict__ deg_in) {
    int e = blockIdx.x * 256 + threadIdx.x;
    if (e < NEDGES) {
        atomicAdd(&deg_out[src[e]], 1.0f);
        atomicAdd(&deg_in[dst[e]], 1.0f);
    }
}

// in-place: d[i] = rsqrt(max(d[i],1))
__global__ __launch_bounds__(256) void k_rsqrt(float* __restrict__ d, int n) {
    int i = blockIdx.x * 256 + threadIdx.x;
    if (i < n) d[i] = rsqrtf(fmaxf(d[i], 1.0f));
}

// ---------------- GEMM1: T1 = (X @ W1) * rs_out[row]  (WMMA f32) ----------------
// One block = 8 waves = one 16-row M tile; wave w handles N tile w (8*16 = 128 cols).

__global__ __launch_bounds__(256) void k_gemm1(const float* __restrict__ X,
                                               const float* __restrict__ W,
                                               const float* __restrict__ rs_out,
                                               float* __restrict__ T1) {
    __shared__ float As[16][132];   // +4 pad: conflict-free column reads
    const int tid = threadIdx.x;
    const int m0  = blockIdx.x * 16;

    // cooperative load of 16x128 A tile (2 x float4 per thread)
    {
        int f = tid * 4;
        int row = f >> 7, col = f & 127;
        float4 v = *(const float4*)&X[(m0 + row) * FIN + col];
        *(float4*)&As[row][col] = v;
        f += 1024;
        row = f >> 7; col = f & 127;
        v = *(const float4*)&X[(m0 + row) * FIN + col];
        *(float4*)&As[row][col] = v;
    }
    __syncthreads();

    const int r    = tid & 15;
    const int half = (tid >> 4) & 1;
    const int n0   = (tid >> 5) * 16;

    v8f acc = {};
    #pragma unroll
    for (int kk = 0; kk < FIN; kk += 4) {
        const int k0 = kk + 2 * half;
        v2f a, b;
        a.x = As[r][k0];
        a.y = As[r][k0 + 1];
        b.x = W[k0 * FHID + n0 + r];
        b.y = W[(k0 + 1) * FHID + n0 + r];
        acc = __builtin_amdgcn_wmma_f32_16x16x4_f32(false, a, false, b,
                                                    (short)0, acc, false, false);
    }
    #pragma unroll
    for (int v = 0; v < 8; ++v) {
        const int row = m0 + v + 8 * half;
        T1[row * FHID + n0 + r] = acc[v] * rs_out[row];
    }
}

// ---------------- edge scatter, 128-wide: agg1[dst] += t1[src] ----------------
// one wave per edge; each lane moves a float4 (4 atomics)

__global__ __launch_bounds__(256) void k_edge128(const int* __restrict__ src,
                                                 const int* __restrict__ dst,
                                                 const float* __restrict__ t1,
                                                 float* __restrict__ agg1) {
    const int e = blockIdx.x * 8 + (threadIdx.x >> 5);
    if (e >= NEDGES) return;
    const int lane = threadIdx.x & 31;
    const int s = src[e], d = dst[e];
    const float4 v = *(const float4*)&t1[s * FHID + lane * 4];
    float* ap = &agg1[d * FHID + lane * 4];
    atomicAdd(ap + 0, v.x);
    atomicAdd(ap + 1, v.y);
    atomicAdd(ap + 2, v.z);
    atomicAdd(ap + 3, v.w);
}

// ---------------- GEMM2: T2 = (relu(agg1*rs_in + b1) @ W2) * rs_out ----------------
// h1 never materialized: fused into A-fragment load. N = 16 -> one N tile per wave,
// wave w of block handles M tile (blockIdx*8 + w).

__global__ __launch_bounds__(256) void k_gemm2(const float* __restrict__ AGG,
                                               const float* __restrict__ b1,
                                               const float* __restrict__ W2,
                                               const float* __restrict__ rs_in,
                                               const float* __restrict__ rs_out,
                                               float* __restrict__ T2) {
    const int tid   = threadIdx.x;
    const int mtile = blockIdx.x * 8 + (tid >> 5);
    const int m0    = mtile * 16;
    if (m0 >= NNODES) return;               // wave-uniform skip (EXEC stays full)

    const int r    = tid & 15;
    const int half = (tid >> 4) & 1;
    const float rsin = rs_in[m0 + r];
    const float* __restrict__ arow = &AGG[(m0 + r) * FHID];

    v8f acc = {};
    #pragma unroll
    for (int kk = 0; kk < FHID; kk += 4) {
        const int k0 = kk + 2 * half;
        v2f a, b;
        a.x = fmaxf(arow[k0]     * rsin + b1[k0],     0.0f);
        a.y = fmaxf(arow[k0 + 1] * rsin + b1[k0 + 1], 0.0f);
        b.x = W2[k0 * FOUT + r];
        b.y = W2[(k0 + 1) * FOUT + r];
        acc = __builtin_amdgcn_wmma_f32_16x16x4_f32(false, a, false, b,
                                                    (short)0, acc, false, false);
    }
    #pragma unroll
    for (int v = 0; v < 8; ++v) {
        const int row = m0 + v + 8 * half;
        T2[row * FOUT + r] = acc[v] * rs_out[row];
    }
}

// ---------------- edge scatter, 16-wide: agg2[dst] += t2[src] ----------------
// 4 lanes per edge, each lane a float4

__global__ __launch_bounds__(256) void k_edge16(const int* __restrict__ src,
                                                const int* __restrict__ dst,
                                                const float* __restrict__ t2,
                                                float* __restrict__ agg2) {
    const int e = blockIdx.x * 64 + (threadIdx.x >> 2);
    if (e >= NEDGES) return;
    const int q = threadIdx.x & 3;
    const int s = src[e], d = dst[e];
    const float4 v = *(const float4*)&t2[s * FOUT + q * 4];
    float* ap = &agg2[d * FOUT + q * 4];
    atomicAdd(ap + 0, v.x);
    atomicAdd(ap + 1, v.y);
    atomicAdd(ap + 2, v.z);
    atomicAdd(ap + 3, v.w);
}

// ---------------- pooling: gsum[g] += agg2[node]*rs_in + b2; gcount[g] += 1 --------

__global__ __launch_bounds__(256) void k_pool(const float* __restrict__ agg2,
                                              const float* __restrict__ rs_in,
                                              const float* __restrict__ b2,
                                              const int* __restrict__ gid,
                                              float* __restrict__ gsum,
                                              float* __restrict__ gcount) {
    const int i = blockIdx.x * 256 + threadIdx.x;
    if (i >= NNODES * FOUT) return;
    const int node = i >> 4;
    const int f    = i & 15;
    const int g    = gid[node];
    const float h  = agg2[i] * rs_in[node] + b2[f];
    atomicAdd(&gsum[g * FOUT + f], h);
    if (f == 0) atomicAdd(&gcount[g], 1.0f);
}

__global__ __launch_bounds__(256) void k_final(const float* __restrict__ gsum,
                                               const float* __restrict__ gcount,
                                               float* __restrict__ out) {
    const int i = blockIdx.x * 256 + threadIdx.x;
    if (i >= NGRAPHS * FOUT) return;
    out[i] = gsum[i] / fmaxf(gcount[i >> 4], 1.0f);
}

// ---------------- host-side launch ----------------

extern "C" void kernel_launch(void* const* d_in, const int* in_sizes, int n_in,
                              void* d_out, int out_size, void* d_ws, size_t ws_size,
                              hipStream_t stream) {
    const float* x   = (const float*)d_in[0];   // [50000,128]
    const float* W1  = (const float*)d_in[1];   // [128,128]
    const float* b1  = (const float*)d_in[2];   // [128]
    const float* W2  = (const float*)d_in[3];   // [128,16]
    const float* b2  = (const float*)d_in[4];   // [16]
    const int*   src = (const int*)d_in[5];     // [800000]
    const int*   dst = (const int*)d_in[6];     // [800000]
    const int*   gid = (const int*)d_in[7];     // [50000]
    float* out = (float*)d_out;                 // [64,16]

    // workspace layout (floats)
    float* ws      = (float*)d_ws;
    float* deg_out = ws;                        // 50000   (-> rs_out after k_rsqrt)
    float* deg_in  = deg_out + NNODES;          // 50000   (-> rs_in)
    float* agg1    = deg_in  + NNODES;          // 6,400,000
    float* agg2    = agg1 + (size_t)NNODES*FHID;// 800,000
    float* gsum    = agg2 + (size_t)NNODES*FOUT;// 1024
    float* gcount  = gsum + NGRAPHS * FOUT;     // 64
    float* t1      = gcount + NGRAPHS;          // 6,400,000
    float* t2      = t1 + (size_t)NNODES*FHID;  // 800,000

    // 1) zero accumulated buffers (deg..gcount are contiguous)
    const int nzero = 2*NNODES + NNODES*FHID + NNODES*FOUT + NGRAPHS*FOUT + NGRAPHS;
    k_zero<<<(nzero + 255) / 256, 256, 0, stream>>>(deg_out, nzero);

    // 2) degrees, then convert to rsqrt(max(deg,1)) in place
    k_degree<<<(NEDGES + 255) / 256, 256, 0, stream>>>(src, dst, deg_out, deg_in);
    k_rsqrt<<<(2 * NNODES + 255) / 256, 256, 0, stream>>>(deg_out, 2 * NNODES);

    // 3) layer 1: GEMM-first (WMMA f32), then 128-wide edge scatter
    k_gemm1<<<NNODES / 16, 256, 0, stream>>>(x, W1, deg_out, t1);
    k_edge128<<<NEDGES / 8, 256, 0, stream>>>(src, dst, t1, agg1);

    // 4) layer 2: fused relu/bias/norm into A-load, GEMM (WMMA f32), 16-wide scatter
    k_gemm2<<<(NNODES / 16 + 7) / 8, 256, 0, stream>>>(agg1, b1, W2, deg_in, deg_out, t2);
    k_edge16<<<NEDGES / 64, 256, 0, stream>>>(src, dst, t2, agg2);

    // 5) per-graph mean pooling
    k_pool<<<(NNODES * FOUT + 255) / 256, 256, 0, stream>>>(agg2, deg_in, b2, gid, gsum, gcount);
    k_final<<<(NGRAPHS * FOUT + 255) / 256, 256, 0, stream>>>(gsum, gcount, out);
}